// CPModule_9019431321787
// MI455X (gfx1250) — compile-verified
//
#include <hip/hip_runtime.h>
#include <hip/hip_bf16.h>

// ---------------------------------------------------------------------------
// Problem constants (from the reference): BS=2, T=4, FEAT=64, H=W=32, K=8
// ---------------------------------------------------------------------------
#define BS    2
#define T_    4
#define FEAT  64
#define HW    1024            // 32*32
#define THW   4096            // T_*HW
#define KNN   8
#define CHUNK 512             // sim columns per LDS chunk in gram/topk kernel
#define CSTR  516             // padded LDS row stride: 16B-aligned rows,
                              // lane stride 2064B == 16 mod 256 -> b128 scan
                              // by 16 lanes hits disjoint 4-bank groups
#define BIGF  3.0e38f

typedef float v2f __attribute__((ext_vector_type(2)));
typedef float v8f __attribute__((ext_vector_type(8)));

// ---------------------------------------------------------------------------
// Kernel 1: transpose x[b][t][f][p] -> xi[(b*THW+i)][f]  (i = t*HW + p)
//           and compute sq[b*THW+i] = sum_f xi^2
// Grid: BS*T_*(HW/64) = 128 blocks of 256 threads. LDS 64x65 transpose tile.
// ---------------------------------------------------------------------------
__global__ __launch_bounds__(256) void pack_kernel(const float* __restrict__ x,
                                                   float* __restrict__ xi,
                                                   float* __restrict__ sq) {
  __shared__ float tile[64][65];
  const int blk = blockIdx.x;
  const int pt  = blk & 15;        // which 64-wide p tile
  const int bt  = blk >> 4;        // b*T_ + t
  const int p0  = pt * 64;
  const int tid = threadIdx.x;

  const float* src = x + (size_t)bt * FEAT * HW + p0;
#pragma unroll
  for (int rep = 0; rep < 16; ++rep) {
    int lin = tid + rep * 256;
    int f = lin >> 6, p = lin & 63;
    tile[f][p] = src[(size_t)f * HW + p];   // coalesced in p
  }
  __syncthreads();

  if (tid < 64) {
    float s = 0.f;
#pragma unroll
    for (int f = 0; f < 64; ++f) { float v = tile[f][tid]; s += v * v; }
    sq[(size_t)bt * HW + p0 + tid] = s;     // b*THW + t*HW + p == bt*HW + p
  }

  float* dst = xi + ((size_t)bt * HW + p0) * FEAT;
#pragma unroll
  for (int rep = 0; rep < 16; ++rep) {
    int lin = tid + rep * 256;
    int p = lin >> 6, f = lin & 63;
    dst[(size_t)p * FEAT + f] = tile[f][p]; // coalesced in f
  }
}

// ---------------------------------------------------------------------------
// Stable sorted-insert into a length-8 (d2 ascending, index ascending on
// ties) list: matches jax.lax.top_k which prefers lower indices on ties.
// ---------------------------------------------------------------------------
__device__ __forceinline__ void topk_insert(float v, int gcol,
                                            float (&bestd)[KNN],
                                            int (&besti)[KNN]) {
  if (v < bestd[KNN - 1]) {                 // strict '<' => stable on ties
    int p = KNN - 1;
    while (p > 0 && v < bestd[p - 1]) {
      bestd[p] = bestd[p - 1];
      besti[p] = besti[p - 1];
      --p;
    }
    bestd[p] = v;
    besti[p] = gcol;
  }
}

// ---------------------------------------------------------------------------
// Kernel 2: fused Gram (fp32 WMMA) + same-frame mask + top-8 (smallest d2).
// One block = one (batch, 16-row strip). 8 waves; each wave owns 64 columns
// per 512-column chunk (4 tiles of 16). sim tile lives only in LDS.
// A-fragments preloaded once; B-fragments preloaded per tile as a batch of
// 16 x global_load_b64 (one clause, one wait) before the 16-WMMA burst.
// top-k: lanes 0..15 of the block each scan one row via ds_load_b128.
// ---------------------------------------------------------------------------
__global__ __launch_bounds__(256) void gram_topk_kernel(const float* __restrict__ xi,
                                                        const float* __restrict__ sq,
                                                        int* __restrict__ knn_idx) {
  __shared__ __align__(16) float simLds[16 * CSTR];   // ~33 KB, padded rows

  const int b   = blockIdx.x >> 8;        // THW/16 = 256 strips per batch
  const int r0  = (blockIdx.x & 255) << 4;
  const float* xb  = xi + (size_t)b * THW * FEAT;
  const float* sqb = sq + (size_t)b * THW;

  const int tid   = threadIdx.x;
  const int lane  = tid & 31;
  const int wave  = tid >> 5;
  const int mrow  = lane & 15;            // A row / B col / C column N
  const int khalf = lane >> 4;            // 0: K pair {0,1}, 1: K pair {2,3}

  // Preload A fragments for the 16 query rows: 16 steps of K=4.
  // 32-bit A 16x4 layout: lane(m)<16 holds K={0,1}; lane(m)+16 holds K={2,3}.
  v2f afrag[16];
  {
    const float* arow = xb + (size_t)(r0 + mrow) * FEAT + khalf * 2;
#pragma unroll
    for (int kk = 0; kk < 16; ++kk)
      afrag[kk] = *(const v2f*)(arow + kk * 4);
  }

  // squared norms for this lane's 8 C-matrix rows (M = khalf*8 + v)
  float sqr[8];
#pragma unroll
  for (int v = 0; v < 8; ++v) sqr[v] = sqb[r0 + khalf * 8 + v];

  // running top-8 (only meaningful for tid < 16)
  float bestd[KNN];
  int   besti[KNN];
#pragma unroll
  for (int j = 0; j < KNN; ++j) { bestd[j] = BIGF; besti[j] = 0; }

  for (int chn = 0; chn < THW / CHUNK; ++chn) {
    const int cchunk0 = chn * CHUNK;

    // -------- compute: 4 column tiles of 16 per wave --------
#pragma unroll
    for (int t4 = 0; t4 < 4; ++t4) {
      const int c0 = cchunk0 + wave * 64 + t4 * 16;
      const float* brow = xb + (size_t)(c0 + mrow) * FEAT + khalf * 2;

      // batch all 16 B fragments first -> one load clause + single wait
      v2f bfrag[16];
#pragma unroll
      for (int kk = 0; kk < 16; ++kk)
        bfrag[kk] = *(const v2f*)(brow + kk * 4);

      v8f acc = {};
#pragma unroll
      for (int kk = 0; kk < 16; ++kk) {
        // D = A(16x4) * B(4x16) + C, fp32  ->  v_wmma_f32_16x16x4_f32
        acc = __builtin_amdgcn_wmma_f32_16x16x4_f32(
            /*neg_a=*/false, afrag[kk], /*neg_b=*/false, bfrag[kk],
            /*c_mod=*/(short)0, acc, /*reuse_a=*/false, /*reuse_b=*/false);
      }

      const float sqc = sqb[c0 + mrow];
      const int   fc  = (c0 + mrow) >> 10;     // column frame (hw = 1024)
#pragma unroll
      for (int v = 0; v < 8; ++v) {
        const int m  = khalf * 8 + v;          // C layout: M = v + 8*(lane>=16)
        const int fr = (r0 + m) >> 10;         // row frame
        float d2 = sqr[v] + sqc - 2.0f * acc[v];
        if (fr == fc) d2 = BIGF;               // same-frame mask (== sim -1000)
        simLds[m * CSTR + (c0 - cchunk0) + mrow] = d2;
      }
    }
    __syncthreads();

    // -------- top-8 update: one lane per query row, b128 LDS reads --------
    if (tid < 16) {
      const float4* rowp = (const float4*)&simLds[tid * CSTR];
      for (int j4 = 0; j4 < CHUNK / 4; ++j4) {
        const float4 v = rowp[j4];
        const float worst = bestd[KNN - 1];
        if (v.x < worst || v.y < worst || v.z < worst || v.w < worst) {
          const int gcol = cchunk0 + j4 * 4;
          topk_insert(v.x, gcol + 0, bestd, besti);
          topk_insert(v.y, gcol + 1, bestd, besti);
          topk_insert(v.z, gcol + 2, bestd, besti);
          topk_insert(v.w, gcol + 3, bestd, besti);
        }
      }
    }
    __syncthreads();
  }

  if (tid < 16) {
    int* dst = knn_idx + ((size_t)b * THW + r0 + tid) * KNN;
#pragma unroll
    for (int j = 0; j < KNN; ++j) dst[j] = besti[j];
  }
}

// ---------------------------------------------------------------------------
// Kernel 3: gather neighbors, build [q(64) | nb(64) | disp(3)] input,
// MLP 131->16->8->64 (no activations), max over k, scatter to NCHW output.
// One wave per (b,i); 8 waves / block; weights staged in LDS.
// ---------------------------------------------------------------------------
__global__ __launch_bounds__(256) void mlp_kernel(const float* __restrict__ xi,
                                                  const int* __restrict__ knn_idx,
                                                  const float* __restrict__ W1,
                                                  const float* __restrict__ b1,
                                                  const float* __restrict__ W2,
                                                  const float* __restrict__ b2,
                                                  const float* __restrict__ W3,
                                                  const float* __restrict__ b3,
                                                  float* __restrict__ out) {
  __shared__ float sW1[16 * 131];
  __shared__ float sW2[8 * 16];
  __shared__ float sW3[64 * 8];
  __shared__ float sb1[16], sb2[8], sb3[64];
  __shared__ float sm[8][132];    // per-wave MLP input (131 used)
  __shared__ float sh1[8][16];
  __shared__ float sh2[8][8];

  const int tid = threadIdx.x;
  for (int i = tid; i < 16 * 131; i += 256) sW1[i] = W1[i];
  for (int i = tid; i < 8 * 16;  i += 256) sW2[i] = W2[i];
  for (int i = tid; i < 64 * 8;  i += 256) sW3[i] = W3[i];
  if (tid < 16) sb1[tid] = b1[tid];
  if (tid < 8)  sb2[tid] = b2[tid];
  if (tid < 64) sb3[tid] = b3[tid];
  __syncthreads();

  const int lane = tid & 31;
  const int wave = tid >> 5;
  const int gi   = blockIdx.x * 8 + wave;  // exactly BS*THW work items
  const int b    = gi >> 12;               // / THW
  const int i    = gi & (THW - 1);

  const float* q  = xi + ((size_t)b * THW + i) * FEAT;
  const int*   nn = knn_idx + ((size_t)b * THW + i) * KNN;

  // query position terms, torch-faithful: it=((i//h)*w)/t, ih=i%h, iw=0
  const float it_ = (float)((i >> 5) << 5) * 0.25f;   // (i//32)*32 / 4
  const float ih_ = (float)(i & 31);

  float o0 = -BIGF, o1 = -BIGF;
  for (int k = 0; k < KNN; ++k) {
    const int ni = nn[k];
    const float* nf = xi + ((size_t)b * THW + ni) * FEAT;
    sm[wave][lane]       = q[lane];
    sm[wave][lane + 32]  = q[lane + 32];
    sm[wave][64 + lane]  = nf[lane];
    sm[wave][96 + lane]  = nf[lane + 32];
    if (lane == 0) {
      const int pp = ni & (HW - 1);
      sm[wave][128] = (float)(ni >> 10) * 0.25f + it_;  // ct + it
      sm[wave][129] = (float)(pp >> 5) + ih_;           // ch + ih
      sm[wave][130] = (float)(pp & 31);                 // cw + 0
    }
    __syncthreads();

    if (lane < 16) {
      float s = sb1[lane];
      const float* w = &sW1[lane * 131];
      for (int d = 0; d < 131; ++d) s += w[d] * sm[wave][d];
      sh1[wave][lane] = s;
    }
    __syncthreads();

    if (lane < 8) {
      float s = sb2[lane];
      const float* w = &sW2[lane * 16];
#pragma unroll
      for (int d = 0; d < 16; ++d) s += w[d] * sh1[wave][d];
      sh2[wave][lane] = s;
    }
    __syncthreads();

    {
      float s0 = sb3[lane], s1 = sb3[lane + 32];
#pragma unroll
      for (int d = 0; d < 8; ++d) {
        const float h = sh2[wave][d];
        s0 += sW3[lane * 8 + d] * h;
        s1 += sW3[(lane + 32) * 8 + d] * h;
      }
      o0 = fmaxf(o0, s0);
      o1 = fmaxf(o1, s1);
    }
    __syncthreads();     // before next k overwrites sm/sh1/sh2
  }

  // out[b][t][c][p], i = t*HW + p
  const int t = i >> 10, p = i & (HW - 1);
  float* dstb = out + (((size_t)b * T_ + t) * FEAT) * HW + p;
  dstb[(size_t)lane * HW]        = o0;
  dstb[(size_t)(lane + 32) * HW] = o1;
}

// ---------------------------------------------------------------------------
// Host launcher. Inputs (setup_inputs order): x, W1, b1, W2, b2, W3, b3.
// Workspace layout: xi (BS*THW*FEAT f32) | sq (BS*THW f32) | knn (BS*THW*K i32)
// ---------------------------------------------------------------------------
extern "C" void kernel_launch(void* const* d_in, const int* in_sizes, int n_in,
                              void* d_out, int out_size, void* d_ws, size_t ws_size,
                              hipStream_t stream) {
  (void)in_sizes; (void)n_in; (void)out_size; (void)ws_size;

  const float* x  = (const float*)d_in[0];
  const float* W1 = (const float*)d_in[1];
  const float* b1 = (const float*)d_in[2];
  const float* W2 = (const float*)d_in[3];
  const float* b2 = (const float*)d_in[4];
  const float* W3 = (const float*)d_in[5];
  const float* b3 = (const float*)d_in[6];
  float* out = (float*)d_out;

  float* xi  = (float*)d_ws;                       // 524288 floats
  float* sq  = xi + (size_t)BS * THW * FEAT;       // 8192 floats
  int*   knn = (int*)(sq + (size_t)BS * THW);      // 65536 ints  (~2.3 MB total)

  pack_kernel<<<BS * T_ * (HW / 64), 256, 0, stream>>>(x, xi, sq);
  gram_topk_kernel<<<BS * (THW / 16), 256, 0, stream>>>(xi, sq, knn);
  mlp_kernel<<<(BS * THW) / 8, 256, 0, stream>>>(xi, knn, W1, b1, W2, b2, W3, b3, out);
}